// GAT_65867618451487
// MI455X (gfx1250) — compile-verified
//
#include <hip/hip_runtime.h>

typedef float v2f __attribute__((ext_vector_type(2)));
typedef float v8f __attribute__((ext_vector_type(8)));

#define B_  4
#define S_  2000
#define F_  128
#define NG  5
#define G_  400
#define H_  5
#define K_  2

// Cross-half lane swap (lane <-> lane^16) via v_permlanex16_b32 (VALU, no LDS).
__device__ __forceinline__ float swap16(float v) {
    int x = __builtin_amdgcn_permlanex16(__float_as_int(v), __float_as_int(v),
                                         0x76543210u, 0xfedcba98u, false, false);
    return __int_as_float(x);
}

__global__ __launch_bounds__(160) void tiny_mha_fused(
    const float* __restrict__ x,
    const float* __restrict__ Wq, const float* __restrict__ bq,
    const float* __restrict__ Wk, const float* __restrict__ bk,
    const float* __restrict__ Wv, const float* __restrict__ bv,
    const float* __restrict__ Wo, const float* __restrict__ bo,
    float* __restrict__ out)
{
    // row stride 10 floats: &arr[t][h*2] is 8-byte aligned (40t + 8h)
    __shared__ __align__(8) float qs[F_][H_ * K_];
    __shared__ __align__(8) float ks[F_][H_ * K_];
    __shared__ __align__(8) float vs[F_][H_ * K_];
    __shared__ __align__(8) float os[F_][H_ * K_];
    __shared__ __align__(8) float zpad[2];      // zero source for K=2,3 pad lanes

    const int blk = blockIdx.x;
    const int b   = blk / G_;
    const int g   = blk % G_;
    const int tid = threadIdx.x;

    const float scale = 0.70710678118f;   // 1/sqrt(K=2), folded into q

    if (tid == 0) { zpad[0] = 0.0f; zpad[1] = 0.0f; }

    // ---------------- Phase 1: QKV projections (one thread per row f) -------
    if (tid < F_) {
        const int f = tid;
        float xv[NG];
        #pragma unroll
        for (int n = 0; n < NG; ++n)
            xv[n] = x[((size_t)b * S_ + (size_t)(g * NG + n)) * F_ + f];

        #pragma unroll
        for (int h = 0; h < H_; ++h) {
            #pragma unroll
            for (int k = 0; k < K_; ++k) {
                const int hk = h * K_ + k;
                float aq = bq[(g * H_ + h) * K_ + k];
                float ak = bk[(g * H_ + h) * K_ + k];
                float av = bv[(g * H_ + h) * K_ + k];
                #pragma unroll
                for (int n = 0; n < NG; ++n) {
                    const int wi = ((g * NG + n) * H_ + h) * K_ + k;
                    aq = fmaf(xv[n], Wq[wi], aq);
                    ak = fmaf(xv[n], Wk[wi], ak);
                    av = fmaf(xv[n], Wv[wi], av);
                }
                qs[f][hk] = aq * scale;
                ks[f][hk] = ak;
                vs[f][hk] = av;
            }
        }
    }
    __syncthreads();

    // ---------------- Phase 2: one wave32 per head ---------------------------
    // Transposed scores: D = K_tile x Q^T_tile = S^T[t][f] via
    // V_WMMA_F32_16X16X4_F32 (K padded 2->4 through a zeroed-LDS address select).
    // t lands in the VGPR-index/half position -> softmax reduction over t is
    // in-lane; only one cross-half permlanex16 combine per accumulator.
    // No max-subtraction: |scores| << 88, f32 exp cannot overflow.
    {
        const int h    = tid >> 5;       // wave id == head (0..4)
        const int lane = tid & 31;
        const int lo   = lane & 15;
        const int half = lane >> 4;      // lanes 16..31 carry K=2,3 -> zeros
        const int h2   = h * K_;

        // Preload the 8 B (query) tiles once; reused by all 8 t-blocks.
        v2f Bq[8];
        #pragma unroll
        for (int j = 0; j < 8; ++j) {
            const float* qp = half ? &zpad[0] : &qs[16 * j + lo][h2];
            Bq[j] = *(const v2f*)qp;
        }

        float sum[8], o0[8], o1[8];
        #pragma unroll
        for (int j = 0; j < 8; ++j) { sum[j] = 0.f; o0[j] = 0.f; o1[j] = 0.f; }

        for (int i = 0; i < 8; ++i) {
            // A-tile: key rows t = 16i..16i+15 (K0/K1 in lanes 0-15, zeros above)
            const float* kp = half ? &zpad[0] : &ks[16 * i + lo][h2];
            const v2f A = *(const v2f*)kp;

            v8f acc[8];
            #pragma unroll
            for (int j = 0; j < 8; ++j) {
                v8f z = {};
                acc[j] = __builtin_amdgcn_wmma_f32_16x16x4_f32(
                    false, A, false, Bq[j], (short)0, z, false, false);
            }

            // V pairs for this lane's t values: t = 16i + v + 8*half
            // (broadcast loads, immediate offsets v*40 bytes)
            v2f Vv[8];
            #pragma unroll
            for (int v = 0; v < 8; ++v)
                Vv[v] = *(const v2f*)&vs[16 * i + v + 8 * half][h2];

            // exp + fully in-lane accumulation over t (VGPR index v)
            #pragma unroll
            for (int j = 0; j < 8; ++j) {
                #pragma unroll
                for (int v = 0; v < 8; ++v) {
                    const float p = __expf(acc[j][v]);
                    sum[j] += p;
                    o0[j]  = fmaf(p, Vv[v].x, o0[j]);
                    o1[j]  = fmaf(p, Vv[v].y, o1[j]);
                }
            }
        }

        // Combine the two t-halves (lane <-> lane^16), normalize, store.
        #pragma unroll
        for (int j = 0; j < 8; ++j) {
            sum[j] += swap16(sum[j]);
            o0[j]  += swap16(o0[j]);
            o1[j]  += swap16(o1[j]);
        }
        if (half == 0) {
            #pragma unroll
            for (int j = 0; j < 8; ++j) {
                const int f   = 16 * j + lo;
                const float inv = 1.0f / sum[j];
                os[f][h2 + 0] = o0[j] * inv;
                os[f][h2 + 1] = o1[j] * inv;
            }
        }
    }
    __syncthreads();

    // ---------------- Phase 3: output projection + store ---------------------
    if (tid < F_) {
        const int f = tid;
        #pragma unroll
        for (int n = 0; n < NG; ++n) {
            float a = bo[g * NG + n];
            #pragma unroll
            for (int i = 0; i < H_ * K_; ++i)
                a = fmaf(os[f][i], Wo[(g * H_ * K_ + i) * NG + n], a);
            out[((size_t)b * S_ + (size_t)(g * NG + n)) * F_ + f] = a;
        }
    }
}

extern "C" void kernel_launch(void* const* d_in, const int* in_sizes, int n_in,
                              void* d_out, int out_size, void* d_ws, size_t ws_size,
                              hipStream_t stream) {
    (void)in_sizes; (void)n_in; (void)d_ws; (void)ws_size; (void)out_size;
    const float* x  = (const float*)d_in[0];
    const float* Wq = (const float*)d_in[1];
    const float* bq = (const float*)d_in[2];
    const float* Wk = (const float*)d_in[3];
    const float* bk = (const float*)d_in[4];
    const float* Wv = (const float*)d_in[5];
    const float* bv = (const float*)d_in[6];
    const float* Wo = (const float*)d_in[7];
    const float* bo = (const float*)d_in[8];
    float* out = (float*)d_out;

    dim3 grid(B_ * G_);   // 1600 independent (b,g) workgroups
    dim3 block(160);      // 5 wave32s = one per head
    tiny_mha_fused<<<grid, block, 0, stream>>>(x, Wq, bq, Wk, bk, Wv, bv, Wo, bo, out);
}